// IGAB_5368709120597
// MI455X (gfx1250) — compile-verified
//
#include <hip/hip_runtime.h>
#include <hip/hip_bf16.h>

// ---------------- constants (match reference) ----------------
#define Bb   4
#define Cc   64
#define Hh   64
#define Wd   64
#define LL   4096          // Hh*Wd
#define HEADS 8
#define HD   8
#define HF   170
#define DIN  128
#define NN   16
#define RR   4
#define SCH  64            // selective-scan chunk length staged in LDS

typedef float v2f __attribute__((ext_vector_type(2)));
typedef float v8f __attribute__((ext_vector_type(8)));
typedef unsigned int u32x4 __attribute__((ext_vector_type(4)));
typedef int i32x4 __attribute__((ext_vector_type(4)));
typedef int i32x8 __attribute__((ext_vector_type(8)));

#if __has_builtin(__builtin_amdgcn_tensor_load_to_lds) && __has_builtin(__builtin_amdgcn_s_wait_tensorcnt)
#define USE_TDM 1
#else
#define USE_TDM 0
#endif

__device__ __forceinline__ float geluf(float v)     { return 0.5f * v * (1.f + erff(v * 0.70710678118654752f)); }
__device__ __forceinline__ float siluf(float v)     { return v / (1.f + expf(-v)); }
__device__ __forceinline__ float softplusf(float v) { return (v > 20.f) ? v : log1pf(expf(v)); }

#if USE_TDM
// ---------------------------------------------------------------------
// Issue a 2-D TENSOR_LOAD_TO_LDS: tile_d1 rows x tile_d0 f32 elements,
// row stride = stride_elems, optional 1-dword LDS pad every 64 dwords.
// Descriptor bit layout per CDNA5 ISA 8.3/8.4 (D# groups 0,1).
// This toolchain's builtin takes 6 args (g0,g1,g2,g3,g4,cpol).
// ---------------------------------------------------------------------
__device__ __forceinline__ void tdm_load_2d(unsigned lds_byte_addr, const float* gptr,
                                            unsigned tile_d0, unsigned tile_d1,
                                            unsigned long long stride_elems, int pad64)
{
    const unsigned long long ga = (unsigned long long)(size_t)gptr;
    u32x4 g0;
    g0[0] = 1u;                                            // count=1, user mode
    g0[1] = lds_byte_addr;                                 // [63:32]  lds_addr
    g0[2] = (unsigned)(ga & 0xffffffffull);                // [95:64]  global_addr lo
    g0[3] = (unsigned)((ga >> 32) & 0x01ffffffull)         // [120:96] global_addr hi
          | 0x80000000u;                                   // [127:126] type=2 (image)
    const unsigned td0 = (unsigned)LL;                     // tensor_dim0 (in-bounds tiles)
    const unsigned td1 = tile_d1;                          // tensor_dim1
    unsigned cw = (2u << 16);                              // data_size = 4 bytes
    if (pad64) cw |= (1u << 20) | (5u << 22);              // pad_enable, every 64 dwords, +1 dword
    i32x8 g1;
    g1[0] = (int)cw;
    g1[1] = (int)((td0 & 0xffffu) << 16);                  // [79:48] tensor_dim0 (lo16)
    g1[2] = (int)(((td0 >> 16) & 0xffffu) | ((td1 & 0xffffu) << 16)); // dim0 hi | dim1 lo
    g1[3] = (int)(((td1 >> 16) & 0xffffu) | ((tile_d0 & 0xffffu) << 16)); // dim1 hi | tile_dim0
    g1[4] = (int)(tile_d1 & 0xffffu);                      // tile_dim1 | tile_dim2=0
    g1[5] = (int)(unsigned)(stride_elems & 0xffffffffull); // tensor_dim0_stride lo32
    g1[6] = (int)((stride_elems >> 32) & 0xffffull);       // stride hi16 | dim1_stride lo = 0
    g1[7] = 0;
    i32x4 g2 = {0, 0, 0, 0};
    i32x4 g3 = {0, 0, 0, 0};
    i32x8 g4 = {0, 0, 0, 0, 0, 0, 0, 0};
    __builtin_amdgcn_tensor_load_to_lds(g0, g1, g2, g3, g4, 0);
}
#endif

// =====================================================================
// FAST fp32 WMMA GEMM (Co%16==0, Ci%4==0): 16(m) x 32(n) tile per wave,
// guard-free inner loop, V_WMMA_F32_16X16X4_F32.  act: 0 none, 2 gelu
// =====================================================================
__global__ __launch_bounds__(32)
void wmma_gemm_fast(const float* __restrict__ Wm, const float* __restrict__ bias,
                    const float* __restrict__ in, const float* __restrict__ res,
                    float* __restrict__ out, int Co, int Ci,
                    long in_bstride, long out_bstride, int act)
{
    const int lane = threadIdx.x & 31;
    const int n0 = blockIdx.x * 32;
    const int m0 = blockIdx.y * 16;
    const int z  = blockIdx.z;

    const float* inp = in + (long)z * in_bstride;
    float*       op  = out + (long)z * out_bstride;
    const float* rp  = res ? (res + (long)z * out_bstride) : nullptr;

    const int hf = lane >> 4;           // which half-wave
    const int lm = lane & 15;
    const int col0 = n0 + lm, col1 = col0 + 16;
    const float* wrow = Wm + (long)(m0 + lm) * Ci + hf * 2;

    v8f acc0 = {}, acc1 = {};
    #pragma unroll 2
    for (int k0 = 0; k0 < Ci; k0 += 4) {
        const v2f a = *(const v2f*)(wrow + k0);
        const float* bp = inp + (long)(k0 + hf * 2) * LL;
        v2f b0, b1;
        b0.x = bp[col0]; b0.y = bp[LL + col0];
        b1.x = bp[col1]; b1.y = bp[LL + col1];
        if (k0 + 4 < Ci) __builtin_prefetch(bp + 4 * LL + col0, 0, 3);
        acc0 = __builtin_amdgcn_wmma_f32_16x16x4_f32(false, a, false, b0, (short)0, acc0, false, false);
        acc1 = __builtin_amdgcn_wmma_f32_16x16x4_f32(false, a, false, b1, (short)0, acc1, false, false);
    }

    #pragma unroll
    for (int i = 0; i < 8; ++i) {
        const int m = m0 + i + hf * 8;
        float v0 = acc0[i], v1 = acc1[i];
        if (bias) { const float bv = bias[m]; v0 += bv; v1 += bv; }
        if (act == 2) { v0 = geluf(v0); v1 = geluf(v1); }
        if (rp) { v0 += rp[(long)m * LL + col0]; v1 += rp[(long)m * LL + col1]; }
        op[(long)m * LL + col0] = v0;
        op[(long)m * LL + col1] = v1;
    }
}

// =====================================================================
// Generic guarded WMMA GEMM (ragged Co / Ci, per-z weight cycling)
// =====================================================================
__global__ __launch_bounds__(32)
void wmma_gemm_gen(const float* __restrict__ Wm, const float* __restrict__ bias,
                   const float* __restrict__ in, float* __restrict__ out,
                   int Co, int Ci, long in_bstride, long out_bstride,
                   int w_cycle, const float* __restrict__ res)
{
    const int lane = threadIdx.x & 31;
    const int n0 = blockIdx.x * 16;
    const int m0 = blockIdx.y * 16;
    const int z  = blockIdx.z;

    const float* Wp  = Wm + (w_cycle ? (long)(z % w_cycle) * Co * Ci : 0);
    const float* inp = in + (long)z * in_bstride;
    float*       op  = out + (long)z * out_bstride;
    const float* rp  = res ? (res + (long)z * out_bstride) : nullptr;

    const int hf = lane >> 4;
    const int lm = lane & 15;
    const int row = m0 + lm;
    const int col = n0 + lm;
    const bool rv = (row < Co);

    v8f acc = {};
    for (int k0 = 0; k0 < Ci; k0 += 4) {
        const int ka = k0 + hf * 2;
        v2f a, b;
        a.x = (rv && ka     < Ci) ? Wp[(long)row * Ci + ka]     : 0.f;
        a.y = (rv && ka + 1 < Ci) ? Wp[(long)row * Ci + ka + 1] : 0.f;
        b.x = (ka     < Ci) ? inp[(long)ka * LL + col]       : 0.f;
        b.y = (ka + 1 < Ci) ? inp[(long)(ka + 1) * LL + col] : 0.f;
        acc = __builtin_amdgcn_wmma_f32_16x16x4_f32(false, a, false, b,
                                                    (short)0, acc, false, false);
    }

    #pragma unroll
    for (int i = 0; i < 8; ++i) {
        const int m = m0 + i + hf * 8;
        if (m < Co) {
            float v = acc[i];
            if (bias) v += bias[m];
            if (rp)   v += rp[(long)m * LL + col];
            op[(long)m * LL + col] = v;
        }
    }
}

// =====================================================================
// Per-pixel LayerNorm over Cn channels (NCL layout), optional silu-gate.
// =====================================================================
__global__ void ln_chan(const float* __restrict__ in, const float* __restrict__ w,
                        const float* __restrict__ b, const float* __restrict__ gate,
                        float* __restrict__ out, int Cn,
                        long in_bstride, long gate_bstride, long out_bstride)
{
    const long idx = (long)blockIdx.x * blockDim.x + threadIdx.x; // over Bb*LL
    if (idx >= (long)Bb * LL) return;
    const int bb = (int)(idx / LL);
    const int l  = (int)(idx % LL);
    const float* ip = in + (long)bb * in_bstride + l;
    float s = 0.f, ss = 0.f;
    for (int c = 0; c < Cn; ++c) { float v = ip[(long)c * LL]; s += v; ss += v * v; }
    const float mu  = s / (float)Cn;
    const float var = ss / (float)Cn - mu * mu;
    const float inv = rsqrtf(var + 1e-5f);
    const float* gp = gate ? (gate + (long)bb * gate_bstride + l) : nullptr;
    float* op = out + (long)bb * out_bstride + l;
    for (int c = 0; c < Cn; ++c) {
        float v = (ip[(long)c * LL] - mu) * inv * w[c] + b[c];
        if (gp) v *= siluf(gp[(long)c * LL]);
        op[(long)c * LL] = v;
    }
}

// =====================================================================
// Depthwise 3x3, pad 1, optional activation (0 none, 1 silu, 2 gelu)
// =====================================================================
__global__ void dw3x3(const float* __restrict__ in, const float* __restrict__ wt,
                      const float* __restrict__ bias, float* __restrict__ out,
                      int Cdw, long in_bstride, long out_bstride, int act, long total)
{
    const long idx = (long)blockIdx.x * blockDim.x + threadIdx.x;
    if (idx >= total) return;
    const int l = (int)(idx % LL);
    const int c = (int)((idx / LL) % Cdw);
    const int b = (int)(idx / ((long)Cdw * LL));
    const int h = l >> 6, w = l & 63;
    const float* ip = in + (long)b * in_bstride + (long)c * LL;
    const float* wp = wt + c * 9;
    float acc = bias ? bias[c] : 0.f;
    #pragma unroll
    for (int dy = -1; dy <= 1; ++dy) {
        const int hh = h + dy;
        if (hh < 0 || hh >= Hh) continue;
        #pragma unroll
        for (int dx = -1; dx <= 1; ++dx) {
            const int ww = w + dx;
            if (ww < 0 || ww >= Wd) continue;
            acc += ip[hh * Wd + ww] * wp[(dy + 1) * 3 + (dx + 1)];
        }
    }
    if (act == 1) acc = siluf(acc);
    else if (act == 2) acc = geluf(acc);
    out[(long)b * out_bstride + (long)c * LL + l] = acc;
}

// =====================================================================
// L2-normalize 64 rows-per-batch of length LL (in place)
// =====================================================================
__global__ void rownorm(float* __restrict__ p, int ctot)
{
    const int r = blockIdx.x;
    const int b = r >> 6, c = r & 63;
    float* row = p + ((long)b * ctot + c) * LL;
    __shared__ float sm[256];
    float s = 0.f;
    for (int i = threadIdx.x; i < LL; i += 256) { float v = row[i]; s += v * v; }
    sm[threadIdx.x] = s; __syncthreads();
    for (int st = 128; st > 0; st >>= 1) {
        if (threadIdx.x < st) sm[threadIdx.x] += sm[threadIdx.x + st];
        __syncthreads();
    }
    const float inv = 1.f / fmaxf(sqrtf(sm[0]), 1e-12f);
    for (int i = threadIdx.x; i < LL; i += 256) row[i] *= inv;
}

// =====================================================================
// Attention (transposed, 8x8 per head)
// =====================================================================
__global__ void attn_scores(const float* __restrict__ q, const float* __restrict__ k,
                            const float* __restrict__ temp, float* __restrict__ attn)
{
    const int bh = blockIdx.x;
    const int b = bh >> 3, h = bh & 7;
    const int c = threadIdx.x >> 3, d = threadIdx.x & 7;
    const float* qr = q + ((long)b * 64  + h * 8 + c) * LL;
    const float* kr = k + ((long)b * 128 + h * 8 + d) * LL;
    float s = 0.f;
    for (int l = 0; l < LL; ++l) s += qr[l] * kr[l];
    attn[bh * 64 + threadIdx.x] = s * temp[h];
}

__global__ void attn_softmax(float* __restrict__ attn)
{
    const int t = threadIdx.x;
    float* row = attn + t * 8;
    float m = row[0];
    #pragma unroll
    for (int i = 1; i < 8; ++i) m = fmaxf(m, row[i]);
    float s = 0.f; float e[8];
    #pragma unroll
    for (int i = 0; i < 8; ++i) { e[i] = expf(row[i] - m); s += e[i]; }
    const float inv = 1.f / s;
    #pragma unroll
    for (int i = 0; i < 8; ++i) row[i] = e[i] * inv;
}

__global__ void attn_apply(const float* __restrict__ attn, const float* __restrict__ kvd,
                           float* __restrict__ out)
{
    const long idx = (long)blockIdx.x * blockDim.x + threadIdx.x;
    if (idx >= (long)Bb * 64 * LL) return;
    const int l = (int)(idx & 4095);
    const int c = (int)((idx >> 12) & 63);
    const int b = (int)(idx >> 18);
    const int h = c >> 3, cc = c & 7;
    const float* ar = attn + ((b * 8 + h) * 8 + cc) * 8;
    const float* vb = kvd + ((long)b * 128 + 64 + h * 8) * LL + l;
    float s = 0.f;
    #pragma unroll
    for (int d = 0; d < 8; ++d) s += ar[d] * vb[(long)d * LL];
    out[idx] = s;
}

__global__ void gelu_gate(const float* __restrict__ gd, float* __restrict__ gg)
{
    const long idx = (long)blockIdx.x * blockDim.x + threadIdx.x;
    if (idx >= (long)Bb * HF * LL) return;
    const int l = (int)(idx % LL);
    const int c = (int)((idx / LL) % HF);
    const int b = (int)(idx / ((long)HF * LL));
    const float g1 = gd[((long)b * (2 * HF) + c) * LL + l];
    const float g2 = gd[((long)b * (2 * HF) + HF + c) * LL + l];
    gg[idx] = geluf(g1) * g2;
}

__global__ void build_xs(const float* __restrict__ xc, float* __restrict__ xs)
{
    const long idx = (long)blockIdx.x * blockDim.x + threadIdx.x;
    if (idx >= (long)Bb * DIN * LL) return;
    const int l = (int)(idx % LL);
    const int d = (int)((idx / LL) % DIN);
    const int b = (int)(idx / ((long)DIN * LL));
    const float v = xc[idx];
    const int h = l >> 6, w = l & 63;
    const int lt = w * Hh + h;
    float* base = xs + ((long)b * 4) * DIN * LL + (long)d * LL;
    base[0 * DIN * LL + l]             = v;
    base[1 * DIN * LL + lt]            = v;
    base[2 * DIN * LL + (LL - 1 - l)]  = v;
    base[3 * DIN * LL + (LL - 1 - lt)] = v;
}

// =====================================================================
// Selective scan: one block per (b,k), 128 threads = d.
// 64-step chunks staged into LDS by the Tensor Data Mover (wave 0 issues
// tensor_load_to_lds, waits TENSORcnt, barrier) with fallback staging.
// dt-projection + softplus inline; direction-dependent scatter via
// global atomic f32 add into the combined ym.
// =====================================================================
__global__ __launch_bounds__(128)
void scan_kernel(const float* __restrict__ xs, const float* __restrict__ xdbl,
                 const float* __restrict__ dt_w, const float* __restrict__ dt_b,
                 const float* __restrict__ A_logs, const float* __restrict__ Ds,
                 float* __restrict__ ym)
{
    const int z = blockIdx.x;          // b*4+k
    const int b = z >> 2, k = z & 3;
    const int d = threadIdx.x;

    __shared__ float sX[36 * SCH];           // dts|B|C rows, contiguous 64-wide
    __shared__ float sU[DIN * (SCH + 1)];    // u rows, stride 65 (padded)

    float A[NN], h[NN], wdt[RR];
    #pragma unroll
    for (int n = 0; n < NN; ++n) { A[n] = -expf(A_logs[((k * DIN + d) << 4) + n]); h[n] = 0.f; }
    #pragma unroll
    for (int r = 0; r < RR; ++r) wdt[r] = dt_w[((long)k * DIN + d) * RR + r];
    const float bdt = dt_b[k * DIN + d];
    const float Dd  = Ds[k * DIN + d];

    const float* xs_base = xs + (long)z * DIN * LL;
    const float* xd      = xdbl + (long)z * 36 * LL;
    float* ymb = ym + ((long)b * DIN + d) * LL;

    for (int c0 = 0; c0 < LL; c0 += SCH) {
#if USE_TDM
        if (threadIdx.x < 32) {
            tdm_load_2d((unsigned)(size_t)(void*)sX, xd + c0, SCH, 36,
                        (unsigned long long)LL, 0);
            tdm_load_2d((unsigned)(size_t)(void*)sU, xs_base + c0, SCH, DIN,
                        (unsigned long long)LL, 1);
            __builtin_amdgcn_s_wait_tensorcnt(0);
        }
        __syncthreads();
#else
        for (int t = threadIdx.x; t < 36 * SCH; t += 128) {
            const int r = t / SCH, j = t - r * SCH;
            sX[t] = xd[(long)r * LL + c0 + j];
        }
        for (int t = threadIdx.x; t < DIN * SCH; t += 128) {
            const int rr = t / SCH, j = t - rr * SCH;
            sU[rr * (SCH + 1) + j] = xs_base[(long)rr * LL + c0 + j];
        }
        __syncthreads();
#endif
        for (int j = 0; j < SCH; ++j) {
            const int l = c0 + j;
            const float dtp = wdt[0] * sX[j] + wdt[1] * sX[SCH + j] +
                              wdt[2] * sX[2 * SCH + j] + wdt[3] * sX[3 * SCH + j] + bdt;
            const float dt = softplusf(dtp);
            const float u  = sU[d * (SCH + 1) + j];
            const float du = dt * u;
            float y = 0.f;
            #pragma unroll
            for (int n = 0; n < NN; ++n) {
                h[n] = expf(dt * A[n]) * h[n] + du * sX[(4 + n) * SCH + j];
                y += h[n] * sX[(20 + n) * SCH + j];
            }
            y += Dd * u;

            int pos;
            if (k == 0)      pos = l;
            else if (k == 1) pos = (l & 63) * Wd + (l >> 6);
            else if (k == 2) pos = LL - 1 - l;
            else { const int jj = LL - 1 - l; pos = (jj & 63) * Wd + (jj >> 6); }
            atomicAdd(&ymb[pos], y);
        }
        __syncthreads();
    }
}

// =====================================================================
// Host-side orchestration
// =====================================================================
extern "C" void kernel_launch(void* const* d_in, const int* in_sizes, int n_in,
                              void* d_out, int out_size, void* d_ws, size_t ws_size,
                              hipStream_t stream)
{
    (void)in_sizes; (void)n_in; (void)out_size; (void)ws_size;

    const float* x         = (const float*)d_in[0];
    const float* illu      = (const float*)d_in[1];
    const float* q_w       = (const float*)d_in[2];
    const float* qdw_w     = (const float*)d_in[3];
    const float* kv_w      = (const float*)d_in[4];
    const float* kvdw_w    = (const float*)d_in[5];
    const float* proj_w    = (const float*)d_in[6];
    const float* ffn_in_w  = (const float*)d_in[7];
    const float* ffn_dw_w  = (const float*)d_in[8];
    const float* ffn_out_w = (const float*)d_in[9];
    const float* in_proj_w = (const float*)d_in[10];
    const float* conv_w    = (const float*)d_in[11];
    const float* x_proj_w  = (const float*)d_in[12];
    const float* dt_w      = (const float*)d_in[13];
    const float* out_proj_w= (const float*)d_in[14];
    const float* ff1_w     = (const float*)d_in[15];
    const float* ffdw_w    = (const float*)d_in[16];
    const float* ff2_w     = (const float*)d_in[17];
    const float* ifa_ln1_w = (const float*)d_in[18];
    const float* ifa_ln1_b = (const float*)d_in[19];
    const float* q_b       = (const float*)d_in[20];
    const float* qdw_b     = (const float*)d_in[21];
    const float* kv_b      = (const float*)d_in[22];
    const float* kvdw_b    = (const float*)d_in[23];
    const float* temp      = (const float*)d_in[24];
    const float* proj_b    = (const float*)d_in[25];
    const float* ifa_ln2_w = (const float*)d_in[26];
    const float* ifa_ln2_b = (const float*)d_in[27];
    const float* ffn_in_b  = (const float*)d_in[28];
    const float* ffn_dw_b  = (const float*)d_in[29];
    const float* ffn_out_b = (const float*)d_in[30];
    const float* conv_b    = (const float*)d_in[31];
    const float* dt_b      = (const float*)d_in[32];
    const float* A_logs    = (const float*)d_in[33];
    const float* Dsp       = (const float*)d_in[34];
    const float* out_ln_w  = (const float*)d_in[35];
    const float* out_ln_b  = (const float*)d_in[36];
    const float* pn_ln_w   = (const float*)d_in[37];
    const float* pn_ln_b   = (const float*)d_in[38];
    float* outp = (float*)d_out;

    // ---- workspace carve-up (floats) ----
    float* ws = (float*)d_ws;
    size_t off = 0;
    const size_t M1 = (size_t)Bb * Cc * LL;           // 1,048,576 floats
    auto take = [&](size_t n) { float* p = ws + off; off += n; return p; };
    float* xn   = take(M1);                           // chanLN(x); reused as oattn
    float* qbuf = take(M1);
    float* kv1  = take(2 * M1);                       // reused as xc
    float* kvd  = take(2 * M1);
    float* attn = take(4096);
    float* xa   = take(M1);
    float* x2   = take(M1);
    float* g    = take((size_t)Bb * 2 * HF * LL);     // reused as f1
    float* gd   = take((size_t)Bb * 2 * HF * LL);     // reused as f1d
    float* gg   = take((size_t)Bb * HF * LL);
    float* yifa = take(M1);
    float* inp2 = take(4 * M1);                       // (B,256,L): xp | z
    float* xs   = take(8 * M1);                       // (B,4,128,L)
    float* xdbl = take((size_t)16 * 36 * LL);
    float* ym   = take(2 * M1);
    float* ygat = take(2 * M1);
    float* xr   = take(M1);
    float* tb   = take(M1);
    float* oattn = xn;
    float* xc    = kv1;
    float* f1    = g;
    float* f1d   = gd;

    const long sC  = (long)Cc * LL;
    const long s2C = 2 * sC;
    const long s4C = 4 * sC;
    const long sG  = (long)2 * HF * LL;
    const long sHF = (long)HF * LL;

    // 1) xn = chanLN(x)
    ln_chan<<<64, 256, 0, stream>>>(x, ifa_ln1_w, ifa_ln1_b, nullptr, xn, Cc, sC, 0, sC);
    // 2) q path
    wmma_gemm_fast<<<dim3(128, 4, Bb), 32, 0, stream>>>(q_w, q_b, xn, nullptr, tb, Cc, Cc, sC, sC, 0);
    dw3x3<<<(int)((long)Bb * Cc * LL / 256), 256, 0, stream>>>(tb, qdw_w, qdw_b, qbuf, Cc, sC, sC, 0, (long)Bb * Cc * LL);
    // 3) kv path
    wmma_gemm_fast<<<dim3(128, 8, Bb), 32, 0, stream>>>(kv_w, kv_b, illu, nullptr, kv1, 2 * Cc, Cc, sC, s2C, 0);
    dw3x3<<<(int)((long)Bb * 2 * Cc * LL / 256), 256, 0, stream>>>(kv1, kvdw_w, kvdw_b, kvd, 2 * Cc, s2C, s2C, 0, (long)Bb * 2 * Cc * LL);
    // 4) L2-normalize q / k rows
    rownorm<<<Bb * 64, 256, 0, stream>>>(qbuf, 64);
    rownorm<<<Bb * 64, 256, 0, stream>>>(kvd, 128);
    // 5) attention
    attn_scores<<<Bb * HEADS, 64, 0, stream>>>(qbuf, kvd, temp, attn);
    attn_softmax<<<1, 256, 0, stream>>>(attn);
    attn_apply<<<(int)((long)Bb * Cc * LL / 256), 256, 0, stream>>>(attn, kvd, oattn);
    // 6) xa = x + proj(oattn)
    wmma_gemm_fast<<<dim3(128, 4, Bb), 32, 0, stream>>>(proj_w, proj_b, oattn, x, xa, Cc, Cc, sC, sC, 0);
    // 7) x2 = chanLN(xa)
    ln_chan<<<64, 256, 0, stream>>>(xa, ifa_ln2_w, ifa_ln2_b, nullptr, x2, Cc, sC, 0, sC);
    // 8) gated FFN (ragged shapes -> guarded kernel)
    wmma_gemm_gen<<<dim3(256, 22, Bb), 32, 0, stream>>>(ffn_in_w, ffn_in_b, x2, g, 2 * HF, Cc, sC, sG, 0, nullptr);
    dw3x3<<<(int)((long)Bb * 2 * HF * LL / 256), 256, 0, stream>>>(g, ffn_dw_w, ffn_dw_b, gd, 2 * HF, sG, sG, 0, (long)Bb * 2 * HF * LL);
    gelu_gate<<<(int)((long)Bb * HF * LL / 256), 256, 0, stream>>>(gd, gg);
    wmma_gemm_gen<<<dim3(256, 4, Bb), 32, 0, stream>>>(ffn_out_w, ffn_out_b, gg, yifa, Cc, HF, sHF, sC, 0, xa);
    // 9) Mamba entry
    wmma_gemm_fast<<<dim3(128, 16, Bb), 32, 0, stream>>>(in_proj_w, nullptr, yifa, nullptr, inp2, 2 * DIN, Cc, sC, s4C, 0);
    dw3x3<<<(int)((long)Bb * DIN * LL / 256), 256, 0, stream>>>(inp2, conv_w, conv_b, xc, DIN, s4C, s2C, 1, (long)Bb * DIN * LL);
    // 10) 4-direction stack + x_proj (per-direction weights)
    build_xs<<<(int)((long)Bb * DIN * LL / 256), 256, 0, stream>>>(xc, xs);
    wmma_gemm_gen<<<dim3(256, 3, Bb * 4), 32, 0, stream>>>(x_proj_w, nullptr, xs, xdbl, RR + 2 * NN, DIN, (long)DIN * LL, (long)36 * LL, 4, nullptr);
    // 11) selective scan (TDM-staged), scatter-accumulate into ym
    (void)hipMemsetAsync(ym, 0, 2 * M1 * sizeof(float), stream);
    scan_kernel<<<Bb * 4, 128, 0, stream>>>(xs, xdbl, dt_w, dt_b, A_logs, Dsp, ym);
    // 12) ygate = LN(ym) * silu(z)
    ln_chan<<<64, 256, 0, stream>>>(ym, out_ln_w, out_ln_b, inp2 + (long)DIN * LL, ygat, DIN, s2C, s4C, s2C);
    // 13) xr = x + out_proj(ygate)
    wmma_gemm_fast<<<dim3(128, 4, Bb), 32, 0, stream>>>(out_proj_w, nullptr, ygat, x, xr, Cc, DIN, s2C, sC, 0);
    // 14) post-norm MLP, final residual into d_out
    ln_chan<<<64, 256, 0, stream>>>(xr, pn_ln_w, pn_ln_b, nullptr, tb, Cc, sC, 0, sC);
    wmma_gemm_fast<<<dim3(128, 16, Bb), 32, 0, stream>>>(ff1_w, nullptr, tb, nullptr, f1, 4 * Cc, Cc, sC, s4C, 2);
    dw3x3<<<(int)((long)Bb * 4 * Cc * LL / 256), 256, 0, stream>>>(f1, ffdw_w, nullptr, f1d, 4 * Cc, s4C, s4C, 2, (long)Bb * 4 * Cc * LL);
    wmma_gemm_fast<<<dim3(128, 4, Bb), 32, 0, stream>>>(ff2_w, nullptr, f1d, xr, outp, Cc, 4 * Cc, s4C, sC, 0);
}